// cap_adj_11948599017965
// MI455X (gfx1250) — compile-verified
//
#include <hip/hip_runtime.h>
#include <math.h>

typedef __attribute__((ext_vector_type(2))) float v2f;
typedef __attribute__((ext_vector_type(8))) float v8f;

#define B_   32
#define T_   12
#define N_   207
#define D_   64
#define H_   8
#define E_   10
#define BT_  (B_ * T_)           // 384
#define MROWS (BT_ * N_)         // 79488 = 4968 * 16
#define NPAD 208                 // N padded to multiple of 4 (K-dim zero pad)

__device__ __forceinline__ v8f wmma4(v2f a, v2f b, v8f c) {
  // D = A(16x4,f32) * B(4x16,f32) + C(16x16,f32)
  return __builtin_amdgcn_wmma_f32_16x16x4_f32(false, a, false, b, (short)0, c,
                                               false, false);
}

// ---------------------------------------------------------------------------
// Kernel 1: Pcaps = squash(x @ W_p.T + b_p)     [MROWS x 64]
// One wave per 16-row tile; 4 column tiles of 16; K=64 in 16 wmma steps.
// Squash fused via shfl_xor reduction within each 16-lane half-group
// (which holds exactly one output row of the C tile).
// ---------------------------------------------------------------------------
__global__ void __launch_bounds__(256) pcaps_kernel(const float* __restrict__ x,
                                                    const float* __restrict__ wp,
                                                    const float* __restrict__ bp,
                                                    float* __restrict__ pc) {
  const int wave = threadIdx.x >> 5;
  const int lane = threadIdx.x & 31;
  const int half = lane >> 4;
  const int l15  = lane & 15;
  const int rowBase = (blockIdx.x * 8 + wave) * 16;
  const int row = rowBase + l15;

  v8f acc0 = {}, acc1 = {}, acc2 = {}, acc3 = {};
  const float* xr = x + (size_t)row * D_;
#pragma unroll
  for (int kk = 0; kk < 16; ++kk) {
    const int keff = kk * 4 + half * 2;
    v2f a  = *(const v2f*)(xr + keff);
    // B[k][j] = W_p[j][k]  (x @ W_p.T)
    v2f b0 = *(const v2f*)(wp + ( 0 + l15) * D_ + keff);
    v2f b1 = *(const v2f*)(wp + (16 + l15) * D_ + keff);
    v2f b2 = *(const v2f*)(wp + (32 + l15) * D_ + keff);
    v2f b3 = *(const v2f*)(wp + (48 + l15) * D_ + keff);
    acc0 = wmma4(a, b0, acc0);
    acc1 = wmma4(a, b1, acc1);
    acc2 = wmma4(a, b2, acc2);
    acc3 = wmma4(a, b3, acc3);
  }
  const float bias0 = bp[ 0 + l15];
  const float bias1 = bp[16 + l15];
  const float bias2 = bp[32 + l15];
  const float bias3 = bp[48 + l15];
#pragma unroll
  for (int r = 0; r < 8; ++r) {
    acc0[r] += bias0; acc1[r] += bias1; acc2[r] += bias2; acc3[r] += bias3;
  }
  // squash each row of 64 (row M = r + half*8, columns striped over 16 lanes)
#pragma unroll
  for (int r = 0; r < 8; ++r) {
    float ss = acc0[r]*acc0[r] + acc1[r]*acc1[r] + acc2[r]*acc2[r] + acc3[r]*acc3[r];
    ss += __shfl_xor(ss, 1, 32);
    ss += __shfl_xor(ss, 2, 32);
    ss += __shfl_xor(ss, 4, 32);
    ss += __shfl_xor(ss, 8, 32);           // reduces within each 16-lane group
    const float sc = (ss / (1.0f + ss)) / (sqrtf(ss) + 1e-8f);
    float* o = pc + (size_t)(rowBase + r + half * 8) * D_;
    o[ 0 + l15] = acc0[r] * sc;
    o[16 + l15] = acc1[r] * sc;
    o[32 + l15] = acc2[r] * sc;
    o[48 + l15] = acc3[r] * sc;
  }
}

// ---------------------------------------------------------------------------
// Kernel 2: per-(b,t) dynamic routing, fully in LDS.
//   dadj  = teb . adj                      [8, 207]
//   test1 = softmax_H(dadj) @ P            [8, 64]   (WMMA, K=208 zero-padded)
//   s     = squash(test1)
//   loop 3x:  c = softmax_H(b)
//             v = squash(s * (c @ P))      (u_hat factorization: no 163MB tensor)
//             b += v @ P^T                 (WMMA, 13 col tiles of 16)
//   out   = softmax_H(b + dadj)
// ---------------------------------------------------------------------------
__global__ void __launch_bounds__(256) routing_kernel(const float* __restrict__ teb,
                                                      const float* __restrict__ adj,
                                                      const float* __restrict__ pc,
                                                      float* __restrict__ out) {
  extern __shared__ float smem[];
  float* Ps   = smem;                    // [NPAD][64]  P, row 207 zeroed
  float* dadj = Ps   + NPAD * D_;        // [8][NPAD]
  float* cmat = dadj + H_ * NPAD;        // [8][NPAD]  softmax buffer (pad col = 0)
  float* blog = cmat + H_ * NPAD;        // [8][NPAD]  routing logits
  float* t1   = blog + H_ * NPAD;        // [8][64]    c @ P
  float* svec = t1   + H_ * D_;          // [8][64]    squash(test1)
  float* vvec = svec + H_ * D_;          // [8][64]    v

  const int tid  = threadIdx.x;
  const int bt   = blockIdx.x;
  const int wave = tid >> 5;
  const int lane = tid & 31;
  const int half = lane >> 4;
  const int l15  = lane & 15;
  const float* Pg = pc + (size_t)bt * N_ * D_;

  float tv[E_];
#pragma unroll
  for (int e = 0; e < E_; ++e) tv[e] = teb[bt * E_ + e];

  // stage P into LDS (float4), zero the K-pad row 207
  {
    float4* Ps4 = (float4*)Ps;
    const float4* Pg4 = (const float4*)Pg;
    for (int i = tid; i < NPAD * (D_ / 4); i += 256) {
      const int n = i >> 4;
      Ps4[i] = (n < N_) ? Pg4[i] : make_float4(0.f, 0.f, 0.f, 0.f);
    }
  }
  // dadj[h][n] = sum_e teb[e] * adj[e,h,n]
  for (int i = tid; i < H_ * N_; i += 256) {
    const int h = i / N_, n = i - h * N_;
    float s = 0.f;
#pragma unroll
    for (int e = 0; e < E_; ++e) s += tv[e] * adj[(e * H_ + h) * N_ + n];
    dadj[h * NPAD + n] = s;
  }
  for (int i = tid; i < H_ * NPAD; i += 256) blog[i] = 0.f;
  if (tid < H_) dadj[tid * NPAD + N_] = 0.f;
  __syncthreads();

  // GEMM (c/softmax @ P): A = cmat [16(pad)x208], B = Ps [208x64] -> t1 [8x64]
  auto gemm_cP = [&]() {
    if (wave < 4) {
      const int jt = wave * 16 + l15;
      v8f acc = {};
      for (int kk = 0; kk < NPAD / 4; ++kk) {
        const int keff = kk * 4 + half * 2;
        v2f a = {0.f, 0.f};
        if (l15 < H_) a = *(const v2f*)(cmat + l15 * NPAD + keff);
        v2f b;
        b.x = Ps[keff * D_ + jt];
        b.y = Ps[(keff + 1) * D_ + jt];
        acc = wmma4(a, b, acc);
      }
      if (half == 0) {
#pragma unroll
        for (int r = 0; r < 8; ++r) t1[r * D_ + jt] = acc[r];
      }
    }
  };

  // column-wise softmax over H: src -> cmat, pad col -> 0
  auto softmaxH = [&](const float* src) {
    for (int n = tid; n < NPAD; n += 256) {
      if (n < N_) {
        float m = src[n];
#pragma unroll
        for (int h = 1; h < H_; ++h) m = fmaxf(m, src[h * NPAD + n]);
        float ex[H_], sum = 0.f;
#pragma unroll
        for (int h = 0; h < H_; ++h) { ex[h] = __expf(src[h * NPAD + n] - m); sum += ex[h]; }
        const float inv = 1.0f / sum;
#pragma unroll
        for (int h = 0; h < H_; ++h) cmat[h * NPAD + n] = ex[h] * inv;
      } else {
#pragma unroll
        for (int h = 0; h < H_; ++h) cmat[h * NPAD + n] = 0.f;
      }
    }
  };

  // test1 = softmax_H(dadj) @ P ; s = squash(test1)
  softmaxH(dadj);
  __syncthreads();
  gemm_cP();
  __syncthreads();
  if (tid < H_) {
    float ss = 0.f;
    for (int d = 0; d < D_; ++d) { const float v = t1[tid * D_ + d]; ss += v * v; }
    const float sc = (ss / (1.f + ss)) / (sqrtf(ss) + 1e-8f);
    for (int d = 0; d < D_; ++d) svec[tid * D_ + d] = t1[tid * D_ + d] * sc;
  }
  __syncthreads();

  for (int it = 0; it < 3; ++it) {
    softmaxH(blog);                      // c
    __syncthreads();
    gemm_cP();                           // t1 = c @ P
    __syncthreads();
    if (tid < H_) {                      // v = squash(s * t1)
      float ss = 0.f;
      for (int d = 0; d < D_; ++d) {
        const float v = svec[tid * D_ + d] * t1[tid * D_ + d];
        vvec[tid * D_ + d] = v;
        ss += v * v;
      }
      const float sc = (ss / (1.f + ss)) / (sqrtf(ss) + 1e-8f);
      for (int d = 0; d < D_; ++d) vvec[tid * D_ + d] *= sc;
    }
    __syncthreads();
    // blog += v @ P^T : A = vvec [16(pad)x64], B[k][j] = Ps[j][k]; 13 col tiles
    for (int t = wave; t < 13; t += 8) {
      const int jt = t * 16 + l15;       // jt==207 reads zero pad row, store masked
      v8f acc = {};
#pragma unroll
      for (int kk = 0; kk < 16; ++kk) {
        const int keff = kk * 4 + half * 2;
        v2f a = {0.f, 0.f};
        if (l15 < H_) a = *(const v2f*)(vvec + l15 * D_ + keff);
        v2f b = *(const v2f*)(Ps + jt * D_ + keff);
        acc = wmma4(a, b, acc);
      }
      if (half == 0 && jt < N_) {
#pragma unroll
        for (int r = 0; r < 8; ++r) blog[r * NPAD + jt] += acc[r];
      }
    }
    __syncthreads();
  }

  // out = softmax_H(blog + dadj)
  for (int n = tid; n < N_; n += 256) {
    float vals[H_];
    float m = -3.402823466e38f;
#pragma unroll
    for (int h = 0; h < H_; ++h) {
      vals[h] = blog[h * NPAD + n] + dadj[h * NPAD + n];
      m = fmaxf(m, vals[h]);
    }
    float sum = 0.f;
#pragma unroll
    for (int h = 0; h < H_; ++h) { vals[h] = __expf(vals[h] - m); sum += vals[h]; }
    const float inv = 1.0f / sum;
#pragma unroll
    for (int h = 0; h < H_; ++h)
      out[((size_t)(bt * H_ + h)) * N_ + n] = vals[h] * inv;
  }
}

// ---------------------------------------------------------------------------
extern "C" void kernel_launch(void* const* d_in, const int* in_sizes, int n_in,
                              void* d_out, int out_size, void* d_ws, size_t ws_size,
                              hipStream_t stream) {
  const float* x   = (const float*)d_in[0];
  const float* teb = (const float*)d_in[1];
  const float* wp  = (const float*)d_in[2];
  const float* bp  = (const float*)d_in[3];
  const float* adj = (const float*)d_in[4];
  // d_in[5] = num_route (==3, fixed in reference; routing loop hardcoded)

  float* pcaps = (float*)d_ws;  // MROWS*64 floats (~19.4 MB)

  pcaps_kernel<<<MROWS / 16 / 8, 256, 0, stream>>>(x, wp, bp, pcaps);

  const size_t shmem =
      (size_t)(NPAD * D_ + 3 * H_ * NPAD + 3 * H_ * D_) * sizeof(float);  // ~79 KB
  routing_kernel<<<BT_, 256, shmem, stream>>>(teb, adj, pcaps, (float*)d_out);
}